// HalfPlane_82669530513756
// MI455X (gfx1250) — compile-verified
//
#include <hip/hip_runtime.h>
#include <math.h>

// Geometry constants (kernel_size=[25,25,25], over_sampling=2)
//   NZ=NY=NX=25, NUM_ANCHOR=35, K padded to 36 for WMMA (K=4 steps x 9)
//   rows R = B*C*NZ = 64*32*25 = 51200, pixels = 625 (39 full 16-wide tiles + 1)

typedef __attribute__((ext_vector_type(2))) float v2f;
typedef __attribute__((ext_vector_type(8))) float v8f;

#define R_TOTAL   51200     // B*C*NZ
#define BC_TOTAL  2048      // B*C
#define NANCH     35
#define KPAD      36
#define NPIX      625
#define NTILES_N  39        // 39*16 = 624 pixels via WMMA; pixel 624 via tail kernel
#define MTILES    3200      // 51200 / 16

// ---------------------------------------------------------------------------
// Kernel 1: per-(b,c) prep. abs -> pin [z=12,a=0]=1 -> compute volume sum via
// anchor column-weights W[a] (closed-form geometry) -> write (25,36) row block
// of psf/sum into workspace (K-padded with zeros).
// ---------------------------------------------------------------------------
__global__ void psf_prep_kernel(const float* __restrict__ params,
                                float* __restrict__ a_scaled) {
    __shared__ float s[NANCH * 25];   // 875 anchors for this (b,c)
    __shared__ float sW[NANCH];       // interpolation-matrix column sums
    __shared__ float sred[256];

    const int bc  = blockIdx.x;
    const int tid = threadIdx.x;
    const float* src = params + (size_t)bc * (NANCH * 25);

    for (int i = tid; i < NANCH * 25; i += 256) s[i] = fabsf(src[i]);
    if (tid < NANCH) sW[tid] = 0.0f;
    __syncthreads();
    if (tid == 0) s[12 * NANCH + 0] = 1.0f;   // center_one pin
    __syncthreads();

    // W[a] = sum over 625 pixels of that pixel's weight on anchor a
    for (int p = tid; p < NPIX; p += 256) {
        int y = p / 25, x = p - y * 25;
        float dx = (float)x - 12.0f, dy = (float)y - 12.0f;
        float rP = sqrtf(dx * dx + dy * dy);
        float fi = floorf(rP * 2.0f);
        int   i1 = (int)fi;
        float d1 = rP * 2.0f - fi;            // weight on i1+1
        atomicAdd(&sW[i1],     1.0f - d1);    // ds_add_f32
        atomicAdd(&sW[i1 + 1], d1);
    }
    __syncthreads();

    // total volume sum = sum_z sum_a s[z,a]*W[a]
    float part = 0.0f;
    for (int i = tid; i < NANCH * 25; i += 256) part += s[i] * sW[i % NANCH];
    sred[tid] = part;
    __syncthreads();
    for (int off = 128; off > 0; off >>= 1) {
        if (tid < off) sred[tid] += sred[tid + off];
        __syncthreads();
    }
    const float inv = 1.0f / sred[0];

    // write K-padded scaled rows: (25 rows) x (36 cols), col 35 = 0
    float* dst = a_scaled + (size_t)bc * 25 * KPAD;
    for (int i = tid; i < 25 * KPAD; i += 256) {
        int z = i / KPAD, a = i - z * KPAD;
        dst[i] = (a < NANCH) ? s[z * NANCH + a] * inv : 0.0f;
    }
}

// ---------------------------------------------------------------------------
// Kernel 2: GEMM  out(51200,625) = A(51200,36) x M(36,625) via
// v_wmma_f32_16x16x4_f32, one 16x16 tile per wave, K chained over 9 steps.
// M (interpolation weights) is generated in registers from the radius formula.
// Per ISA 7.12.2 (32-bit A 16x4): lane L holds A[m=L&15, k0+2*(L>>4)+{0,1}];
// B symmetric: B[k0+2*(L>>4)+{0,1}, n=L&15]. D: lane L, vgpr j ->
// (m = j + 8*(L>>4), n = L&15).
// ---------------------------------------------------------------------------
__global__ void psf_wmma_kernel(const float* __restrict__ a_scaled,
                                float* __restrict__ out) {
    const int wave  = (int)((blockIdx.x * blockDim.x + threadIdx.x) >> 5);
    const int lane  = threadIdx.x & 31;
    const int mtile = wave / NTILES_N;
    const int ntile = wave - mtile * NTILES_N;
    if (mtile >= MTILES) return;            // wave-uniform; EXEC stays all-1s

    const int half = lane >> 4;             // 0 or 1 (selects K pair)
    const int l15  = lane & 15;

    // B-operand pixel for this lane: n = l15
    const int p = ntile * 16 + l15;
    const int y = p / 25, x = p - y * 25;
    const float dx = (float)x - 12.0f, dy = (float)y - 12.0f;
    const float rP = sqrtf(dx * dx + dy * dy);
    const float fi = floorf(rP * 2.0f);
    const int   i1 = (int)fi;
    const float d1 = rP * 2.0f - fi;        // weight on anchor i1+1
    const float w0 = 1.0f - d1;             // weight on anchor i1

    const float* arow = a_scaled + (size_t)(mtile * 16 + l15) * KPAD;

    v8f acc = {0.f, 0.f, 0.f, 0.f, 0.f, 0.f, 0.f, 0.f};
#pragma unroll
    for (int k0 = 0; k0 < KPAD; k0 += 4) {
        const int kb = k0 + half * 2;
        // A[m, kb], A[m, kb+1]  (8-byte aligned: KPAD even, kb even)
        v2f a = *(const v2f*)(arow + kb);
        // B[kb, n], B[kb+1, n] from closed form (2 nonzeros per column)
        float b0 = (kb     == i1) ? w0 : ((kb     == i1 + 1) ? d1 : 0.0f);
        float b1 = (kb + 1 == i1) ? w0 : ((kb + 1 == i1 + 1) ? d1 : 0.0f);
        v2f b = {b0, b1};
        // 8 args: (neg_a, A, neg_b, B, c_mod, C, reuse_a, reuse_b)
        acc = __builtin_amdgcn_wmma_f32_16x16x4_f32(
            false, a, false, b, (short)0, acc, false, false);
    }

    // Store D: acc[j] -> row (mtile*16 + half*8 + j), col p. Already normalized.
    const int rbase = mtile * 16 + half * 8;
#pragma unroll
    for (int j = 0; j < 8; ++j) {
        out[(size_t)(rbase + j) * NPIX + p] = acc[j];
    }
}

// ---------------------------------------------------------------------------
// Kernel 3: leftover pixel column p=624 (y=24,x=24) for all rows.
// ---------------------------------------------------------------------------
__global__ void psf_tail_kernel(const float* __restrict__ a_scaled,
                                float* __restrict__ out) {
    const int r = blockIdx.x * blockDim.x + threadIdx.x;
    if (r >= R_TOTAL) return;
    const float rP = sqrtf(288.0f);          // dx=dy=12
    const float fi = floorf(rP * 2.0f);
    const int   i1 = (int)fi;
    const float d1 = rP * 2.0f - fi;
    const float* arow = a_scaled + (size_t)r * KPAD;
    out[(size_t)r * NPIX + 624] = arow[i1] * (1.0f - d1) + arow[i1 + 1] * d1;
}

// ---------------------------------------------------------------------------
extern "C" void kernel_launch(void* const* d_in, const int* in_sizes, int n_in,
                              void* d_out, int out_size, void* d_ws, size_t ws_size,
                              hipStream_t stream) {
    (void)in_sizes; (void)n_in; (void)out_size; (void)ws_size;
    const float* params = (const float*)d_in[0];
    float* out = (float*)d_out;
    float* a_scaled = (float*)d_ws;          // 51200 * 36 * 4 = 7.37 MB

    psf_prep_kernel<<<BC_TOTAL, 256, 0, stream>>>(params, a_scaled);

    // 3200 M-tiles * 39 N-tiles = 124800 waves; 8 waves/block -> 15600 blocks
    psf_wmma_kernel<<<(MTILES * NTILES_N + 7) / 8, 256, 0, stream>>>(a_scaled, out);

    psf_tail_kernel<<<(R_TOTAL + 255) / 256, 256, 0, stream>>>(a_scaled, out);
}